// NTXentLoss_42941083025747
// MI455X (gfx1250) — compile-verified
//
#include <hip/hip_runtime.h>

typedef float v2f __attribute__((ext_vector_type(2)));
typedef float v4f __attribute__((ext_vector_type(4)));
typedef float v8f __attribute__((ext_vector_type(8)));

#define NROWS 8192
#define DDIM  128
#define BSTRIDE 132   // padded LDS row stride (floats): 16B-aligned rows, spread banks

// ---------------- Kernel 1: reps = concat(zjs, zis); row-normalize ----------------
__global__ __launch_bounds__(256) void ntx_normalize(const float* __restrict__ zis,
                                                     const float* __restrict__ zjs,
                                                     float* __restrict__ repsN) {
    const int wave = threadIdx.x >> 5;
    const int lane = threadIdx.x & 31;
    const int row  = blockIdx.x * 8 + wave;           // 0..8191, one wave per row
    const float* src = (row < 4096) ? (zjs + (size_t)row * DDIM)
                                    : (zis + (size_t)(row - 4096) * DDIM);
    v4f v = *(const v4f*)(src + lane * 4);
    float ss = v.x * v.x + v.y * v.y + v.z * v.z + v.w * v.w;
    ss += __shfl_xor(ss, 1);
    ss += __shfl_xor(ss, 2);
    ss += __shfl_xor(ss, 4);
    ss += __shfl_xor(ss, 8);
    ss += __shfl_xor(ss, 16);
    float nrm = fmaxf(sqrtf(ss), 1e-8f);
    float inv = 1.0f / nrm;
    v4f o = { v.x * inv, v.y * inv, v.z * inv, v.w * inv };
    *(v4f*)(repsN + (size_t)row * DDIM + lane * 4) = o;
}

// ---------------- Kernel 2: fused GEMM (fp32 WMMA) + exp + masked row-sums ----------------
// grid.x = 128 row-blocks (4 waves * 16 rows = 64 rows each)
// grid.y = 4 column chunks of 2048 cols (each chunk == one parity block)
__global__ __launch_bounds__(128) void ntx_main(const float* __restrict__ repsN,
                                                float* __restrict__ partial,   // [4][8192]
                                                float* __restrict__ diagw) {   // [8192]
    __shared__ float bt[16 * BSTRIDE];

    const int wave = threadIdx.x >> 5;
    const int lane = threadIdx.x & 31;
    const int m    = lane & 15;          // row/col within tile (per half-wave)
    const int g    = lane >> 4;          // K-group select for 16x16x4 layout

    const int rowTile = blockIdx.x * 4 + wave;   // 0..511
    const int rowBase = rowTile * 16;

    // Preload A fragments for all 32 K-steps: lane L holds row rowBase+(L&15),
    // K = 4s + 2*(L>>4) + {0,1}  -> contiguous float2 per step.
    v2f areg[32];
    {
        const float* ap = repsN + (size_t)(rowBase + m) * DDIM + 2 * g;
#pragma unroll
        for (int s = 0; s < 32; ++s)
            areg[s] = *(const v2f*)(ap + 4 * s);
    }

    v8f acc = { 0.f, 0.f, 0.f, 0.f, 0.f, 0.f, 0.f, 0.f };
    const int colTile0 = blockIdx.y * 128;

    for (int t = 0; t < 128; ++t) {
        const int colTile = colTile0 + t;
        const int colBase = colTile * 16;

        __syncthreads();
        // Stage B tile (16 rows x 128 floats) coalesced into LDS: 512 float4s / 128 threads
#pragma unroll
        for (int k = 0; k < 4; ++k) {
            int i  = threadIdx.x + k * 128;
            int r  = i >> 5;
            int c4 = i & 31;
            v4f v = *(const v4f*)(repsN + (size_t)(colBase + r) * DDIM + 4 * c4);
            *(v4f*)(&bt[r * BSTRIDE + 4 * c4]) = v;
        }
        __syncthreads();

        v8f sim = { 0.f, 0.f, 0.f, 0.f, 0.f, 0.f, 0.f, 0.f };
        const float* brow = &bt[m * BSTRIDE + 2 * g];
#pragma unroll
        for (int s = 0; s < 32; ++s) {
            v2f b = *(const v2f*)(brow + 4 * s);
            sim = __builtin_amdgcn_wmma_f32_16x16x4_f32(
                false, areg[s], false, b, (short)0, sim, false, false);
        }

        float e[8];
#pragma unroll
        for (int r = 0; r < 8; ++r) {
            float ev = __expf(2.0f * sim[r]);   // 1/TEMP = 2
            e[r] = ev;
            acc[r] += ev;
        }

        if (colTile == rowTile) {
            // diag elements: VGPR r at lane r (row rowBase+r) and lane r+24 (row rowBase+8+r)
#pragma unroll
            for (int r = 0; r < 8; ++r) {
                if (lane == r)      diagw[rowBase + r]     = e[r];
                if (lane == r + 24) diagw[rowBase + 8 + r] = e[r];
            }
        }
    }

    // Row-sum reduce: 16-lane butterfly within each half-wave; lane0 -> row rowBase+r,
    // lane16 -> row rowBase+8+r. Unique writes per (chunk,row): deterministic.
    float* chunkOut = partial + (size_t)blockIdx.y * NROWS;
#pragma unroll
    for (int r = 0; r < 8; ++r) {
        float v = acc[r];
        v += __shfl_xor(v, 1);
        v += __shfl_xor(v, 2);
        v += __shfl_xor(v, 4);
        v += __shfl_xor(v, 8);
        if (lane == 0)       chunkOut[rowBase + r]     = v;
        else if (lane == 16) chunkOut[rowBase + 8 + r] = v;
    }
}

// ---------------- Kernel 3: combine chunks by parity, loss, mean ----------------
__global__ __launch_bounds__(1024) void ntx_finalize(const float* __restrict__ partial,
                                                     const float* __restrict__ diagw,
                                                     float* __restrict__ out) {
    __shared__ float red[1024];
    const int tid = threadIdx.x;
    float lsum = 0.f;
    for (int i = tid; i < NROWS; i += 1024) {
        int p = (i >> 11) & 1;   // row parity block
        float pos = partial[(size_t)p * NROWS + i]
                  + partial[(size_t)(p + 2) * NROWS + i]
                  - diagw[i];
        float neg = partial[(size_t)(1 - p) * NROWS + i]
                  + partial[(size_t)(3 - p) * NROWS + i];
        lsum += logf(neg) - logf(pos);   // -log(pos/neg)
    }
    red[tid] = lsum;
    __syncthreads();
    for (int s = 512; s > 0; s >>= 1) {
        if (tid < s) red[tid] += red[tid + s];
        __syncthreads();
    }
    if (tid == 0) out[0] = red[0] / (float)NROWS;
}

extern "C" void kernel_launch(void* const* d_in, const int* in_sizes, int n_in,
                              void* d_out, int out_size, void* d_ws, size_t ws_size,
                              hipStream_t stream) {
    (void)in_sizes; (void)n_in; (void)out_size; (void)ws_size;
    const float* zis = (const float*)d_in[0];
    const float* zjs = (const float*)d_in[1];

    float* ws      = (float*)d_ws;
    float* repsN   = ws;                                  // 8192*128 floats (4 MB)
    float* partial = repsN + (size_t)NROWS * DDIM;        // 4*8192 floats
    float* diagw   = partial + 4 * (size_t)NROWS;         // 8192 floats

    ntx_normalize<<<dim3(1024), dim3(256), 0, stream>>>(zis, zjs, repsN);
    ntx_main<<<dim3(128, 4), dim3(128), 0, stream>>>(repsN, partial, diagw);
    ntx_finalize<<<dim3(1), dim3(1024), 0, stream>>>(partial, diagw, (float*)d_out);
}